// SpanPooler_29918742184516
// MI455X (gfx1250) — compile-verified
//
#include <hip/hip_runtime.h>
#include <hip/hip_bf16.h>

typedef __attribute__((ext_vector_type(2))) float v2f;
typedef __attribute__((ext_vector_type(8))) float v8f;

// Span mean pooling: out[b, :] = mean over rows [start_b, end_b) of x[b, :, :].
// One wave per 16-column strip; V_WMMA_F32_16X16X4_F32 with an all-ones A
// matrix reduces 4 span rows x 16 columns per instruction in full f32.
__global__ __launch_bounds__(256) void span_mean_wmma_kernel(
    const float* __restrict__ x,     // [B, S, D] f32 (L=1 folded away)
    const int*   __restrict__ spans, // [B, 2] (start, end)
    float*       __restrict__ out,   // [B, D] f32
    int S, int D) {
  const int b    = blockIdx.x;
  const int wave = threadIdx.x >> 5;   // 8 waves per block
  const int lane = threadIdx.x & 31;
  const int col  = lane & 15;          // B-matrix column N = lane & 15
  const int half = lane >> 4;          // which pair of K rows this lane feeds

  const int strip = blockIdx.y * 8 + wave;      // 16-column strip index
  if (strip * 16 >= D) return;                  // wave-uniform exit
  const int c0 = strip * 16 + col;

  const int start = spans[2 * b];
  const int end   = spans[2 * b + 1];
  int cnt = end - start;
  if (cnt < 1) cnt = 1;
  const float inv = 1.0f / (float)cnt;

  const float* base = x + (size_t)b * (size_t)S * (size_t)D;

  v2f ones;
  ones[0] = 1.0f;
  ones[1] = 1.0f;

  v8f acc = {};  // C/D accumulator, 16x16 f32 (every row identical w/ ones-A)

  int r = start;
  // Main loop: 4 span rows per WMMA.
  // lanes 0-15 supply rows r, r+1 (col = lane); lanes 16-31 rows r+2, r+3.
  for (; r + 4 <= end; r += 4) {
    const float* p = base + (size_t)(r + 2 * half) * (size_t)D + c0;
    __builtin_prefetch(p + 4 * (size_t)D, 0, 1);  // next 4-row tile -> global_prefetch
    v2f bm;
    bm[0] = p[0];
    bm[1] = p[D];
    acc = __builtin_amdgcn_wmma_f32_16x16x4_f32(
        /*neg_a=*/false, ones, /*neg_b=*/false, bm,
        /*c_mod=*/(short)0, acc, /*reuse_a=*/false, /*reuse_b=*/false);
  }
  // Tail: 1-3 remaining rows, zero-fill out-of-span rows (exact for a sum).
  if (r < end) {
    const int r0 = r + 2 * half;
    const int r1 = r0 + 1;
    v2f bm;
    bm[0] = (r0 < end) ? base[(size_t)r0 * (size_t)D + c0] : 0.0f;
    bm[1] = (r1 < end) ? base[(size_t)r1 * (size_t)D + c0] : 0.0f;
    acc = __builtin_amdgcn_wmma_f32_16x16x4_f32(
        false, ones, false, bm, (short)0, acc, false, false);
  }

  // D row M=0 lives in acc[0] of lanes 0-15 (N = lane). Lanes 16-31 hold the
  // duplicate M=8 row; skip them.
  if (half == 0) {
    out[(size_t)b * (size_t)D + c0] = acc[0] * inv;
  }
}

extern "C" void kernel_launch(void* const* d_in, const int* in_sizes, int n_in,
                              void* d_out, int out_size, void* d_ws, size_t ws_size,
                              hipStream_t stream) {
  const float* x     = (const float*)d_in[0];  // [1, B, S, D] f32
  const int*   spans = (const int*)d_in[1];    // [B, 2] int32
  float*       out   = (float*)d_out;          // [B, D] f32

  const int B = in_sizes[1] / 2;               // 64
  const int D = out_size / B;                  // 768
  const int S = (int)((long long)in_sizes[0] / ((long long)B * D));  // 2048 (L=1)

  const int strips = (D + 15) / 16;            // 48
  dim3 grid(B, (strips + 7) / 8);              // 64 x 6 = 384 workgroups
  span_mean_wmma_kernel<<<grid, 256, 0, stream>>>(x, spans, out, S, D);
}